// QuantumAttentionLayer_65481071407973
// MI455X (gfx1250) — compile-verified
//
#include <hip/hip_runtime.h>
#include <math.h>
#include <stdint.h>

typedef __attribute__((ext_vector_type(2))) float v2f;
typedef __attribute__((ext_vector_type(8))) float v8f;

// Problem constants (fixed by reference: B=2,S=2048,E=1024,H=16,Dk=64)
constexpr int kB = 2;
constexpr int kS = 2048;
constexpr int kE = 1024;
constexpr int kH = 16;
constexpr int kD = 64;
constexpr int kM = kB * kS;  // 4096 token rows

// GEMM tiling: block 128x128, 8 waves (4M x 2N), wave tile 32x64.
constexpr int BM = 128;
constexpr int BN = 128;
constexpr int BK = 32;
constexpr int XPAD = 44;  // 176B row stride: 16B aligned, conflict-free b64 frags

// Attention tiling
constexpr int AQ = 128;  // queries per block
constexpr int AK = 64;   // keys per LDS stage
constexpr int KPAD = 68; // 272B stride (16B multiple)
constexpr int VPAD = 72; // 288B stride (16B multiple)
constexpr int PPAD = 68;

// ---- CDNA5 async global->LDS copy path (ASYNCcnt), with safe fallback ----
#if defined(__HIP_DEVICE_COMPILE__) &&                                   \
    __has_builtin(__builtin_amdgcn_global_load_async_to_lds_b128) &&     \
    __has_builtin(__builtin_amdgcn_s_wait_asynccnt)
#define USE_ASYNC_LDS 1
#else
#define USE_ASYNC_LDS 0
#endif

__device__ __forceinline__ void cp16_to_lds(void* lds, const void* g) {
#if USE_ASYNC_LDS
  // Builtin prototype (from compiler diagnostic): param0 is
  // 'int __vector_size__(16) __device__ *' (AS1); LDS side is AS3.
  typedef int v4i __attribute__((vector_size(16)));
  typedef __attribute__((address_space(1))) v4i g_v4i;
  typedef __attribute__((address_space(3))) v4i l_v4i;
  // flat->LDS addrspacecast == low 32 bits of the generic pointer.
  __builtin_amdgcn_global_load_async_to_lds_b128(
      (g_v4i*)(uintptr_t)g, (l_v4i*)(uint32_t)(uintptr_t)lds, 0, 0);
#else
  *(float4*)lds = *(const float4*)g;
#endif
}

__device__ __forceinline__ void cp_commit_wait() {
#if USE_ASYNC_LDS
  __builtin_amdgcn_s_wait_asynccnt(0);
#endif
}

__device__ __forceinline__ v8f wmma4(v2f a, v2f b, v8f c) {
  // D = A(16x4 f32) * B(4x16 f32) + C(16x16 f32)
  return __builtin_amdgcn_wmma_f32_16x16x4_f32(false, a, false, b, (short)0, c,
                                               false, false);
}

__device__ __forceinline__ v8f v8zero() {
  v8f z = {0.f, 0.f, 0.f, 0.f, 0.f, 0.f, 0.f, 0.f};
  return z;
}

// ---- GEMM: C = X[M,K] * W[N,K]^T (torch Linear, bias=false), K = N = kE ----
// MODE 0: scatter output into (B,H,S,Dk) layout (for q/k/v).
// MODE 1: row-major [M, kE] output (final projection).

__device__ __forceinline__ void stage_gemm(const float* __restrict__ X,
                                           const float* __restrict__ W,
                                           float (&Xs)[BM][XPAD],
                                           float (&Ws)[BN][XPAD], int m0,
                                           int n0, int k0, int tid) {
#pragma unroll
  for (int it = 0; it < 4; ++it) {
    int idx = it * 256 + tid;  // 1024 float4 slots: 128 rows x 8
    int row = idx >> 3, q = idx & 7;
    cp16_to_lds(&Xs[row][q * 4], X + (size_t)(m0 + row) * kE + k0 + q * 4);
  }
#pragma unroll
  for (int it = 0; it < 4; ++it) {
    int idx = it * 256 + tid;
    int row = idx >> 3, q = idx & 7;
    cp16_to_lds(&Ws[row][q * 4], W + (size_t)(n0 + row) * kE + k0 + q * 4);
  }
}

template <int MODE>
__global__ __launch_bounds__(256) void gemm_nt_kernel(
    const float* __restrict__ X, const float* __restrict__ W,
    float* __restrict__ Out) {
  __shared__ __align__(16) float Xs[2][BM][XPAD];
  __shared__ __align__(16) float Ws[2][BN][XPAD];

  const int tid  = threadIdx.x;
  const int lane = tid & 31;
  const int wave = tid >> 5;
  const int wm   = wave & 3;   // 4 waves along M -> 32 rows each
  const int wn   = wave >> 2;  // 2 waves along N -> 64 cols each
  const int m0   = blockIdx.x * BM;
  const int n0   = blockIdx.y * BN;
  const int r    = lane & 15;
  const int half = lane >> 4;

  v8f acc[2][4];
#pragma unroll
  for (int i = 0; i < 2; ++i)
#pragma unroll
    for (int j = 0; j < 4; ++j) acc[i][j] = v8zero();

  // Prologue: stage first K-slab.
  stage_gemm(X, W, Xs[0], Ws[0], m0, n0, 0, tid);
  cp_commit_wait();
  __syncthreads();

  for (int k0 = 0; k0 < kE; k0 += BK) {
    const int buf = (k0 >> 5) & 1;
    // Kick off the next slab's async copies while this slab computes.
    if (k0 + BK < kE)
      stage_gemm(X, W, Xs[buf ^ 1], Ws[buf ^ 1], m0, n0, k0 + BK, tid);

#pragma unroll
    for (int kk = 0; kk < BK; kk += 4) {
      const int kf = kk + 2 * half;
      v2f a0 = {Xs[buf][wm * 32 + r][kf], Xs[buf][wm * 32 + r][kf + 1]};
      v2f a1 = {Xs[buf][wm * 32 + 16 + r][kf],
                Xs[buf][wm * 32 + 16 + r][kf + 1]};
      v2f b[4];
#pragma unroll
      for (int t = 0; t < 4; ++t) {
        v2f bt = {Ws[buf][wn * 64 + t * 16 + r][kf],
                  Ws[buf][wn * 64 + t * 16 + r][kf + 1]};
        b[t] = bt;
      }
#pragma unroll
      for (int t = 0; t < 4; ++t) {
        acc[0][t] = wmma4(a0, b[t], acc[0][t]);
        acc[1][t] = wmma4(a1, b[t], acc[1][t]);
      }
    }
    cp_commit_wait();
    __syncthreads();
  }

#pragma unroll
  for (int mt = 0; mt < 2; ++mt)
#pragma unroll
    for (int nt = 0; nt < 4; ++nt)
#pragma unroll
      for (int i = 0; i < 8; ++i) {
        int m = m0 + wm * 32 + mt * 16 + i + 8 * half;
        int n = n0 + wn * 64 + nt * 16 + r;
        float val = acc[mt][nt][i];
        if (MODE == 1) {
          Out[(size_t)m * kE + n] = val;
        } else {
          int b = m >> 11, s = m & (kS - 1);
          int h = n >> 6, d = n & (kD - 1);
          Out[(((size_t)(b * kH + h) * kS) + s) * kD + d] = val;
        }
      }
}

// quantum[b,h,s,:] = cumprod(cos(q[b,h,s,:])); one row (64 floats) per thread.
__global__ __launch_bounds__(256) void quantum_kernel(
    const float* __restrict__ q, float* __restrict__ quant) {
  int row = blockIdx.x * blockDim.x + threadIdx.x;  // 0 .. B*H*S-1
  const float4* src = (const float4*)(q + (size_t)row * kD);
  float4* dst = (float4*)(quant + (size_t)row * kD);
  float p = 1.0f;
#pragma unroll
  for (int i = 0; i < kD / 4; ++i) {
    float4 c = src[i];
    float4 o;
    p *= __cosf(c.x); o.x = p;
    p *= __cosf(c.y); o.y = p;
    p *= __cosf(c.z); o.z = p;
    p *= __cosf(c.w); o.w = p;
    dst[i] = o;
  }
}

// ---- Flash attention per (b,h), fused with quantum blend ----
// grid = (S/AQ, B*H), 256 threads = 8 waves, each wave owns 16 query rows.
__global__ __launch_bounds__(256) void attn_kernel(
    const float* __restrict__ Q, const float* __restrict__ Km,
    const float* __restrict__ Vm, const float* __restrict__ quant,
    const float* __restrict__ qw_ptr, float* __restrict__ ctx) {
  __shared__ __align__(16) float Ks[2][AK][KPAD];
  __shared__ __align__(16) float Vs[2][AK][VPAD];
  __shared__ __align__(16) float Ps[8][16][PPAD];  // per-wave P tile (16 x AK)

  const int tid  = threadIdx.x;
  const int lane = tid & 31;
  const int wave = tid >> 5;
  const int r    = lane & 15;
  const int half = lane >> 4;
  const int bh   = blockIdx.y;
  const int q0   = blockIdx.x * AQ;
  const size_t base = (size_t)bh * kS * kD;
  const float* Qb = Q + base;
  const float* Kb = Km + base;
  const float* Vb = Vm + base;
  const int qrow = q0 + wave * 16;

  // Preload this wave's Q fragments (16x64) into registers.
  v2f af[16];
#pragma unroll
  for (int k4 = 0; k4 < 16; ++k4) {
    const float* p = Qb + (size_t)(qrow + r) * kD + k4 * 4 + 2 * half;
    v2f a = {p[0], p[1]};
    af[k4] = a;
  }

  v8f oacc[4];
  float mrow[8], lrow[8];
#pragma unroll
  for (int t = 0; t < 4; ++t) oacc[t] = v8zero();
#pragma unroll
  for (int i = 0; i < 8; ++i) {
    mrow[i] = -INFINITY;
    lrow[i] = 0.f;
  }

  // Stage helper: 64x64 K tile + 64x64 V tile, b128 async per lane.
  auto stage_kv = [&](int buf, int kb) {
#pragma unroll
    for (int it = 0; it < 4; ++it) {
      int idx = it * 256 + tid;  // 1024 float4 slots: 64 rows x 16
      int row = idx >> 4, q = idx & 15;
      cp16_to_lds(&Ks[buf][row][q * 4],
                  Kb + (size_t)(kb + row) * kD + q * 4);
      cp16_to_lds(&Vs[buf][row][q * 4],
                  Vb + (size_t)(kb + row) * kD + q * 4);
    }
  };

  stage_kv(0, 0);
  cp_commit_wait();
  __syncthreads();

  for (int kb = 0; kb < kS; kb += AK) {
    const int buf = (kb >> 6) & 1;
    if (kb + AK < kS) stage_kv(buf ^ 1, kb + AK);

    // Scores: S(16 x 64) = Q_w (16x64) x K_tile^T, K-loop over d.
    v8f sacc[4];
#pragma unroll
    for (int t = 0; t < 4; ++t) sacc[t] = v8zero();
#pragma unroll
    for (int kk = 0; kk < kD; kk += 4) {
      const int kf = kk + 2 * half;
      v2f a = af[kk >> 2];
#pragma unroll
      for (int t = 0; t < 4; ++t) {
        v2f b = {Ks[buf][t * 16 + r][kf], Ks[buf][t * 16 + r][kf + 1]};
        sacc[t] = wmma4(a, b, sacc[t]);
      }
    }

    // Online softmax. Row m = i + 8*half lives in one 16-lane half.
#pragma unroll
    for (int i = 0; i < 8; ++i) {
      float rm = fmaxf(fmaxf(sacc[0][i], sacc[1][i]),
                       fmaxf(sacc[2][i], sacc[3][i])) * 0.125f;
#pragma unroll
      for (int off = 8; off >= 1; off >>= 1)
        rm = fmaxf(rm, __shfl_xor(rm, off, 32));
      float mnew = fmaxf(mrow[i], rm);
      float corr = __expf(mrow[i] - mnew);
      mrow[i] = mnew;
      lrow[i] *= corr;
#pragma unroll
      for (int t = 0; t < 4; ++t) oacc[t][i] *= corr;
    }

    // P = exp(S/8 - m) -> per-wave LDS tile; accumulate row sums.
#pragma unroll
    for (int i = 0; i < 8; ++i) {
      float rs = 0.f;
#pragma unroll
      for (int t = 0; t < 4; ++t) {
        float pex = __expf(sacc[t][i] * 0.125f - mrow[i]);
        Ps[wave][i + 8 * half][t * 16 + r] = pex;
        rs += pex;
      }
#pragma unroll
      for (int off = 8; off >= 1; off >>= 1) rs += __shfl_xor(rs, off, 32);
      lrow[i] += rs;
    }

    // O += P (16 x AK) x V_tile (AK x 64). LDS per-wave ops are in-order.
#pragma unroll
    for (int kk = 0; kk < AK; kk += 4) {
      const int kf = kk + 2 * half;
      v2f a = {Ps[wave][r][kf], Ps[wave][r][kf + 1]};
#pragma unroll
      for (int t = 0; t < 4; ++t) {
        v2f b = {Vs[buf][kf][t * 16 + r], Vs[buf][kf + 1][t * 16 + r]};
        oacc[t] = wmma4(a, b, oacc[t]);
      }
    }
    cp_commit_wait();
    __syncthreads();
  }

  // Epilogue: normalize, blend with quantum term, write ctx in (B,S,E).
  const float qw = qw_ptr[0];
  const float w = 1.f / (1.f + __expf(-qw));
  const int b = bh >> 4, h = bh & 15;
#pragma unroll
  for (int t = 0; t < 4; ++t)
#pragma unroll
    for (int i = 0; i < 8; ++i) {
      int m = i + 8 * half;
      int s = qrow + m;
      int d = t * 16 + r;
      float cls = oacc[t][i] / lrow[i];
      float qv = quant[base + (size_t)s * kD + d];
      ctx[((size_t)(b * kS + s)) * kE + h * kD + d] = w * qv + (1.f - w) * cls;
    }
}

extern "C" void kernel_launch(void* const* d_in, const int* in_sizes, int n_in,
                              void* d_out, int out_size, void* d_ws,
                              size_t ws_size, hipStream_t stream) {
  const float* x  = (const float*)d_in[0];
  const float* Wq = (const float*)d_in[1];
  const float* Wk = (const float*)d_in[2];
  const float* Wv = (const float*)d_in[3];
  const float* Wo = (const float*)d_in[4];
  // d_in[5] = qparam: provably unused (RZ commutes with Z-string observable).
  const float* qw = (const float*)d_in[6];
  float* out = (float*)d_out;

  const size_t N4 = (size_t)kM * kE;  // elements per (B,H,S,Dk) tensor
  float* q     = (float*)d_ws;
  float* k     = q + N4;
  float* v     = k + N4;
  float* quant = v + N4;
  float* ctx   = quant + N4;

  dim3 gblk(kM / BM, kE / BN);  // 32 x 8
  gemm_nt_kernel<0><<<gblk, 256, 0, stream>>>(x, Wq, q);
  gemm_nt_kernel<0><<<gblk, 256, 0, stream>>>(x, Wk, k);
  gemm_nt_kernel<0><<<gblk, 256, 0, stream>>>(x, Wv, v);
  quantum_kernel<<<(kB * kH * kS) / 256, 256, 0, stream>>>(q, quant);
  attn_kernel<<<dim3(kS / AQ, kB * kH), 256, 0, stream>>>(q, k, v, quant, qw,
                                                          ctx);
  gemm_nt_kernel<1><<<gblk, 256, 0, stream>>>(ctx, Wo, out);
}